// LatentMixer_44616120271417
// MI455X (gfx1250) — compile-verified
//
#include <hip/hip_runtime.h>
#include <hip/hip_bf16.h>
#include <math.h>

// ---------------------------------------------------------------------------
// LatentMixer on MI455X (gfx1250): all GEMM-shaped ops on V_WMMA_F32_16X16X4_F32
// (fp32-exact); big-GEMM LDS staging via double-buffered Tensor Data Mover
// (tensor_load_to_lds + s_wait_tensorcnt) overlapped with WMMA compute;
// k/q column norms fused into the GEMM epilogue. wave32 throughout.
// ---------------------------------------------------------------------------

typedef __attribute__((ext_vector_type(2))) float v2f;
typedef __attribute__((ext_vector_type(8))) float v8f;
typedef __attribute__((ext_vector_type(4))) unsigned int v4u;
typedef __attribute__((ext_vector_type(8))) unsigned int v8u;

#define B_   8
#define C_   512
#define N_   4096     // H*W
#define L_   16
#define H_   8        // heads
#define D_   64       // head dim
#define EPS_ 1e-12f
#define SCALE_ 0.125f // d^-0.5

__device__ __forceinline__ v8f wmma4(v2f a, v2f b, v8f c) {
    // D = A(16x4) * B(4x16) + C(16x16), fp32
    return __builtin_amdgcn_wmma_f32_16x16x4_f32(
        false, a, false, b, (short)0, c, false, false);
}

__device__ __forceinline__ v8f zero8() {
    v8f z = {0.f,0.f,0.f,0.f,0.f,0.f,0.f,0.f};
    return z;
}

// ---------------------------------------------------------------------------
// TDM: 2D tile load global -> LDS with LDS row padding (D# per
// CDNA5 ISA; groups 2/3 NULL -> 2D form).
// ---------------------------------------------------------------------------
__device__ __forceinline__ void tdm_load_2d(unsigned int lds_byte, const void* gptr,
                                            unsigned int tdim0, unsigned int tdim1,
                                            unsigned int stride0,
                                            unsigned int tile0, unsigned int tile1,
                                            unsigned int padi, unsigned int pada) {
    unsigned long long ga = (unsigned long long)gptr;
    v4u g0;
    g0[0] = 1u;                                            // count=1, user mode
    g0[1] = lds_byte;                                      // lds_addr
    g0[2] = (unsigned int)ga;                              // global_addr[31:0]
    g0[3] = (unsigned int)((ga >> 32) & 0x01FFFFFFull)     // global_addr[56:32]
          | (2u << 30);                                    // type=2 (image)
    v8u g1;
    g1[0] = (2u << 16)            // data_size = 4 bytes
          | (1u << 20)            // pad_enable
          | (padi << 22)          // pad_interval
          | (pada << 25);         // pad_amount
    g1[1] = (tdim0 & 0xFFFFu) << 16;                       // tensor_dim0 lo16
    g1[2] = (tdim0 >> 16) | ((tdim1 & 0xFFFFu) << 16);     // dim0 hi16 | dim1 lo16
    g1[3] = (tdim1 >> 16) | (tile0 << 16);                 // dim1 hi16 | tile_dim0
    g1[4] = tile1;                                         // tile_dim1 (tile_dim2=0)
    g1[5] = stride0;                                       // tensor_dim0_stride lo32
    g1[6] = 0u;
    g1[7] = 0u;
    asm volatile("tensor_load_to_lds %0, %1" :: "s"(g0), "s"(g1) : "memory");
}

// ---------------------------------------------------------------------------
// K0: concat [w_kv_x (1024x512); w_q_x (512x512)] -> W1 (1536x512)
// ---------------------------------------------------------------------------
__global__ __launch_bounds__(256) void lm_concat_w(const float* __restrict__ wkv,
                                                   const float* __restrict__ wq,
                                                   float* __restrict__ W1) {
    int i = blockIdx.x * 256 + threadIdx.x;              // < 1536*512
    const int split = 1024 * 512;
    W1[i] = (i < split) ? wkv[i] : wq[i - split];
}

// ---------------------------------------------------------------------------
// K1: ql = w_q_lat @ latents, L2-normalized over d.  qln layout [h][l][64].
// ---------------------------------------------------------------------------
__global__ __launch_bounds__(64) void lm_ql(const float* __restrict__ wql,
                                            const float* __restrict__ lat,
                                            float* __restrict__ qln) {
    int l  = blockIdx.x & 15;
    int hh = blockIdx.x >> 4;
    int dd = threadIdx.x;
    int co = hh * D_ + dd;
    float s = 0.f;
    for (int ci = 0; ci < C_; ++ci)
        s = fmaf(wql[co * C_ + ci], lat[ci * L_ + l], s);
    __shared__ float red[64];
    red[dd] = s * s;
    __syncthreads();
    for (int st = 32; st > 0; st >>= 1) {
        if (dd < st) red[dd] += red[dd + st];
        __syncthreads();
    }
    float rn = 1.f / fmaxf(sqrtf(red[0]), EPS_);
    qln[hh * (L_ * D_) + l * D_ + dd] = s * rn;
}

// ---------------------------------------------------------------------------
// K2/K6: fp32 WMMA GEMM, TDM double-buffered slab staging.
// C[b] = A(MxK) * Bmat[b](KxN) (+bias).  block = 256 threads (8 waves);
// block tile 128(M) x 64(N); K slab 32.  Wave 0 issues next-slab TDM while
// all waves compute on the current slab (TENSORcnt overlap).
// Optional fused epilogue: per-column 1/max(||col||,eps) over each 64-row
// head group, written to rkx (rows<512) / rqx (rows>=1024) — removes the
// separate 134 MB norm pass.
// ---------------------------------------------------------------------------
__global__ __launch_bounds__(256) void lm_gemm_f32(const float* __restrict__ A,
                                                   const float* __restrict__ Bmat,
                                                   float* __restrict__ Cmat,
                                                   const float* __restrict__ bias,
                                                   float* __restrict__ rkx,
                                                   float* __restrict__ rqx,
                                                   int M, int N, int K,
                                                   long long strideB, long long strideC) {
    const int bz = blockIdx.z;
    const float* Bp = Bmat + (long long)bz * strideB;
    float*       Cp = Cmat + (long long)bz * strideC;
    const int n0 = blockIdx.x * 64;
    const int m0 = blockIdx.y * 128;

    __shared__ float As[2][128][34];  // even stride: b64 A-frags, conflict-free
    __shared__ float Bs[2][32][80];   // half-wave rows on disjoint bank groups
    __shared__ float nsum[8][64];     // norm epilogue scratch

    const int tid  = threadIdx.x;
    const int wave = tid >> 5;
    const int lane = tid & 31;
    const int lr   = lane & 15;       // lane%16
    const int lh   = lane >> 4;       // half-wave select

    v8f acc0 = zero8(), acc1 = zero8(), acc2 = zero8(), acc3 = zero8();

    const unsigned int ldsA[2] = { (unsigned int)(uintptr_t)&As[0][0][0],
                                   (unsigned int)(uintptr_t)&As[1][0][0] };
    const unsigned int ldsB[2] = { (unsigned int)(uintptr_t)&Bs[0][0][0],
                                   (unsigned int)(uintptr_t)&Bs[1][0][0] };

    const int nslab = K >> 5;

    // prologue: slab 0 into buffer 0
    if (wave == 0) {
        tdm_load_2d(ldsA[0], A + (long long)m0 * K,
                    (unsigned)K, (unsigned)M, (unsigned)K,
                    32u, 128u, /*padi=*/4u, /*pada=*/1u);   // rows of 34 dwords
        tdm_load_2d(ldsB[0], Bp + n0,
                    (unsigned)N, (unsigned)K, (unsigned)N,
                    64u, 32u, /*padi=*/5u, /*pada=*/15u);   // rows of 80 dwords
        __builtin_amdgcn_s_wait_tensorcnt(0);
    }
    __syncthreads();

    for (int s = 0; s < nslab; ++s) {
        const int cur = s & 1;
        if (wave == 0 && s + 1 < nslab) {
            const int k1 = (s + 1) << 5;
            tdm_load_2d(ldsA[cur ^ 1], A + (long long)m0 * K + k1,
                        (unsigned)K, (unsigned)M, (unsigned)K,
                        32u, 128u, 4u, 1u);
            tdm_load_2d(ldsB[cur ^ 1], Bp + (long long)k1 * N + n0,
                        (unsigned)N, (unsigned)K, (unsigned)N,
                        64u, 32u, 5u, 15u);
        }

        const float (*Asc)[34] = As[cur];
        const float (*Bsc)[80] = Bs[cur];
        const int mrow = wave * 16 + lr;
        #pragma unroll
        for (int kk = 0; kk < 32; kk += 4) {
            v2f a = *(const v2f*)&Asc[mrow][kk + 2 * lh];     // ds_load_b64
            v2f b0, b1, b2, b3;
            b0.x = Bsc[kk + 2 * lh][ 0 + lr]; b0.y = Bsc[kk + 2 * lh + 1][ 0 + lr];
            b1.x = Bsc[kk + 2 * lh][16 + lr]; b1.y = Bsc[kk + 2 * lh + 1][16 + lr];
            b2.x = Bsc[kk + 2 * lh][32 + lr]; b2.y = Bsc[kk + 2 * lh + 1][32 + lr];
            b3.x = Bsc[kk + 2 * lh][48 + lr]; b3.y = Bsc[kk + 2 * lh + 1][48 + lr];
            acc0 = wmma4(a, b0, acc0);
            acc1 = wmma4(a, b1, acc1);
            acc2 = wmma4(a, b2, acc2);
            acc3 = wmma4(a, b3, acc3);
        }

        if (wave == 0) __builtin_amdgcn_s_wait_tensorcnt(0);  // next slab landed
        __syncthreads();
    }

    // store: C/D layout -> VGPR g holds rows g (lanes 0-15) and g+8 (lanes 16-31)
    #pragma unroll
    for (int g = 0; g < 8; ++g) {
        int row = m0 + wave * 16 + g + lh * 8;
        float bv = bias ? bias[row] : 0.f;
        long long base = (long long)row * N + n0 + lr;
        Cp[base +  0] = acc0[g] + bv;
        Cp[base + 16] = acc1[g] + bv;
        Cp[base + 32] = acc2[g] + bv;
        Cp[base + 48] = acc3[g] + bv;
    }

    // fused column-norm epilogue (only for the kv/q producer GEMM)
    if (rkx) {
        float p0 = 0.f, p1 = 0.f, p2 = 0.f, p3 = 0.f;
        #pragma unroll
        for (int g = 0; g < 8; ++g) {
            p0 = fmaf(acc0[g], acc0[g], p0);
            p1 = fmaf(acc1[g], acc1[g], p1);
            p2 = fmaf(acc2[g], acc2[g], p2);
            p3 = fmaf(acc3[g], acc3[g], p3);
        }
        // merge the two half-wave row sets (rows g and g+8) per column
        p0 += __shfl_xor(p0, 16, 32);
        p1 += __shfl_xor(p1, 16, 32);
        p2 += __shfl_xor(p2, 16, 32);
        p3 += __shfl_xor(p3, 16, 32);
        if (lh == 0) {
            nsum[wave][ 0 + lr] = p0;
            nsum[wave][16 + lr] = p1;
            nsum[wave][32 + lr] = p2;
            nsum[wave][48 + lr] = p3;
        }
        __syncthreads();
        if (tid < 128) {
            int grp = tid >> 6, col = tid & 63;     // head group within block tile
            float s = nsum[grp * 4 + 0][col] + nsum[grp * 4 + 1][col]
                    + nsum[grp * 4 + 2][col] + nsum[grp * 4 + 3][col];
            float rn = 1.f / fmaxf(sqrtf(s), EPS_);
            int headrow = m0 + grp * 64;
            if (headrow < 512)                       // kx rows
                rkx[((long long)bz * 8 + (headrow >> 6)) * N + n0 + col] = rn;
            else if (headrow >= 1024)                // qx rows
                rqx[((long long)bz * 8 + ((headrow - 1024) >> 6)) * N + n0 + col] = rn;
            // vx rows (512..1023): no norm needed
        }
    }
}

// ---------------------------------------------------------------------------
// K4a: S[bh][l][n] = (q̂ᵀ k)[l,n] * rk[n] * scale  via WMMA (M=16,K=64)
// ---------------------------------------------------------------------------
__global__ __launch_bounds__(256) void lm_attn1_scores(const float* __restrict__ Y1,
                                                       const float* __restrict__ qln,
                                                       const float* __restrict__ rkx,
                                                       float* __restrict__ Sbuf) {
    int bh = blockIdx.y;
    int b  = bh >> 3, hh = bh & 7;
    __shared__ float qT[16][66];    // [l][d], even row stride -> b64-able
    for (int i = threadIdx.x; i < L_ * D_; i += 256)
        qT[i >> 6][i & 63] = qln[hh * (L_ * D_) + i];
    __syncthreads();

    const int wave = threadIdx.x >> 5, lane = threadIdx.x & 31;
    const int lr = lane & 15, lh = lane >> 4;
    const int ntile = blockIdx.x * 128 + wave * 16;
    const float* kb = Y1 + ((long long)b * 1536 + hh * D_) * N_;

    v8f acc = zero8();
    #pragma unroll
    for (int kk = 0; kk < D_; kk += 4) {
        v2f a = *(const v2f*)&qT[lr][kk + 2 * lh];
        v2f bb;
        int n = ntile + lr;
        bb.x = kb[(long long)(kk + 2 * lh) * N_ + n];
        bb.y = kb[(long long)(kk + 2 * lh + 1) * N_ + n];
        acc = wmma4(a, bb, acc);
    }
    float rk = rkx[bh * N_ + ntile + lr] * SCALE_;  // per-column scale
    float* Sb = Sbuf + (long long)bh * (L_ * N_);
    #pragma unroll
    for (int g = 0; g < 8; ++g)
        Sb[(g + lh * 8) * N_ + ntile + lr] = acc[g] * rk;
}

// ---------------------------------------------------------------------------
// K4b: per row (bh*16+l): rowmax and 1/sum(exp).
// ---------------------------------------------------------------------------
__global__ __launch_bounds__(256) void lm_attn1_stats(const float* __restrict__ Sbuf,
                                                      float* __restrict__ mrow,
                                                      float* __restrict__ rsrow) {
    const float* base = Sbuf + (long long)blockIdx.x * N_;
    __shared__ float red[256];
    int t = threadIdx.x;
    float lm = -3.0e38f;
    for (int i = t; i < N_; i += 256) lm = fmaxf(lm, base[i]);
    red[t] = lm; __syncthreads();
    for (int s = 128; s > 0; s >>= 1) {
        if (t < s) red[t] = fmaxf(red[t], red[t + s]);
        __syncthreads();
    }
    float m = red[0]; __syncthreads();
    float ls = 0.f;
    for (int i = t; i < N_; i += 256) ls += __expf(base[i] - m);
    red[t] = ls; __syncthreads();
    for (int s = 128; s > 0; s >>= 1) {
        if (t < s) red[t] += red[t + s];
        __syncthreads();
    }
    if (t == 0) { mrow[blockIdx.x] = m; rsrow[blockIdx.x] = 1.f / red[0]; }
}

// ---------------------------------------------------------------------------
// K4c: lat_read[d,l] = sum_n P[l,n] * vx[d,n]  via WMMA (M=64,N=16,K=4096),
// P computed on the fly.  grid 64, block 128 (4 waves).
// ---------------------------------------------------------------------------
__global__ __launch_bounds__(128) void lm_attn1_out(const float* __restrict__ Y1,
                                                    const float* __restrict__ Sbuf,
                                                    const float* __restrict__ mrow,
                                                    const float* __restrict__ rsrow,
                                                    float* __restrict__ latr) {
    int bh = blockIdx.x;
    int b  = bh >> 3, hh = bh & 7;
    __shared__ float mS[16], rsS[16];
    if (threadIdx.x < 16) {
        mS[threadIdx.x]  = mrow[bh * 16 + threadIdx.x];
        rsS[threadIdx.x] = rsrow[bh * 16 + threadIdx.x];
    }
    __syncthreads();

    const int wave = threadIdx.x >> 5, lane = threadIdx.x & 31;
    const int lr = lane & 15, lh = lane >> 4;
    const int d0 = wave * 16;
    const float* vb = Y1 + ((long long)b * 1536 + 512 + hh * D_) * N_;
    const float* Sb = Sbuf + (long long)bh * (L_ * N_);
    const float ml = mS[lr], rl = rsS[lr];

    v8f acc = zero8();
    for (int kk = 0; kk < N_; kk += 4) {
        v2f a, bb;
        long long ar = (long long)(d0 + lr) * N_ + kk + 2 * lh;
        a.x = vb[ar]; a.y = vb[ar + 1];
        long long sr = (long long)lr * N_ + kk + 2 * lh;
        bb.x = __expf(Sb[sr] - ml) * rl;
        bb.y = __expf(Sb[sr + 1] - ml) * rl;
        acc = wmma4(a, bb, acc);
    }
    #pragma unroll
    for (int g = 0; g < 8; ++g) {
        int row = b * C_ + hh * D_ + d0 + g + lh * 8;
        latr[(long long)row * L_ + lr] = acc[g];
    }
}

// ---------------------------------------------------------------------------
// K5a: kvl = w_kv_lat @ lat_read  (B,1024,16).
// ---------------------------------------------------------------------------
__global__ __launch_bounds__(256) void lm_kvl(const float* __restrict__ wkvl,
                                              const float* __restrict__ latr,
                                              float* __restrict__ kvl) {
    int idx = blockIdx.x * 256 + threadIdx.x;   // < 8*1024*16
    int b = idx >> 14, r = (idx >> 4) & 1023, l = idx & 15;
    float s = 0.f;
    for (int ci = 0; ci < C_; ++ci)
        s = fmaf(wkvl[r * C_ + ci], latr[(b * C_ + ci) * L_ + l], s);
    kvl[idx] = s;
}

// K5b: rkl[b,h,l] = 1/max(||kl[:,l]||, eps).
__global__ __launch_bounds__(1024) void lm_rkl(const float* __restrict__ kvl,
                                               float* __restrict__ rkl) {
    int t = threadIdx.x;                 // < 1024
    int b = t >> 7, hh = (t >> 4) & 7, l = t & 15;
    float ss = 0.f;
    for (int dd = 0; dd < D_; ++dd) {
        float v = kvl[b * 16384 + (hh * D_ + dd) * L_ + l];
        ss = fmaf(v, v, ss);
    }
    rkl[t] = 1.f / fmaxf(sqrtf(ss), EPS_);
}

// ---------------------------------------------------------------------------
// K5: attn2, fully in-wave per 16-column tile.
//   S2 = qxᵀ kl (WMMA) -> cosine scale -> softmax over L=16 via wave32
//   __shfl_xor(.,16) -> transpose via LDS -> x_write = vl·P2ᵀ (WMMA).
// ---------------------------------------------------------------------------
__global__ __launch_bounds__(256) void lm_attn2(const float* __restrict__ Y1,
                                                const float* __restrict__ kvl,
                                                const float* __restrict__ rqx,
                                                const float* __restrict__ rkl,
                                                float* __restrict__ xw) {
    int bh = blockIdx.y;
    int b  = bh >> 3, hh = bh & 7;
    const int wave = threadIdx.x >> 5, lane = threadIdx.x & 31;
    const int lr = lane & 15, lh = lane >> 4;
    const int ntile = blockIdx.x * 128 + wave * 16;

    const float* qb  = Y1  + ((long long)b * 1536 + 1024 + hh * D_) * N_;
    const float* klb = kvl + b * 16384 + (hh * D_) * L_;
    const float* vlb = kvl + b * 16384 + (512 + hh * D_) * L_;

    // --- S2 = qxᵀ kl ---
    v8f acc = zero8();
    #pragma unroll
    for (int kk = 0; kk < D_; kk += 4) {
        v2f a, bb;
        a.x = qb[(long long)(kk + 2 * lh) * N_ + ntile + lr];
        a.y = qb[(long long)(kk + 2 * lh + 1) * N_ + ntile + lr];
        bb.x = klb[(kk + 2 * lh) * L_ + lr];
        bb.y = klb[(kk + 2 * lh + 1) * L_ + lr];
        acc = wmma4(a, bb, acc);
    }

    // --- scale + per-row softmax over 16 lanes (half-wave groups) ---
    float rl = rkl[bh * L_ + lr] * SCALE_;   // per-column (l)
    float p[8];
    #pragma unroll
    for (int g = 0; g < 8; ++g) {
        float rq = rqx[bh * N_ + ntile + g + lh * 8];  // per-row (n)
        float v = acc[g] * rq * rl;
        float mx = v;
        for (int off = 8; off; off >>= 1) mx = fmaxf(mx, __shfl_xor(mx, off, 16));
        float e = __expf(v - mx);
        float sm = e;
        for (int off = 8; off; off >>= 1) sm += __shfl_xor(sm, off, 16);
        p[g] = e / sm;
    }

    // --- transpose P2 through per-wave LDS slot ---
    __shared__ float pt[8][16][17];
    #pragma unroll
    for (int g = 0; g < 8; ++g) pt[wave][g + lh * 8][lr] = p[g];  // [n][l]
    __syncthreads();

    // --- x_write[d, ntile..+15] = vl(64x16) * P2ᵀ(16x16) ---
    #pragma unroll
    for (int t = 0; t < 4; ++t) {
        v8f o = zero8();
        #pragma unroll
        for (int kk = 0; kk < L_; kk += 4) {
            v2f a, bb;
            a.x = vlb[(t * 16 + lr) * L_ + kk + 2 * lh];
            a.y = vlb[(t * 16 + lr) * L_ + kk + 2 * lh + 1];
            bb.x = pt[wave][lr][kk + 2 * lh];
            bb.y = pt[wave][lr][kk + 2 * lh + 1];
            o = wmma4(a, bb, o);
        }
        #pragma unroll
        for (int g = 0; g < 8; ++g) {
            int row = b * C_ + hh * D_ + t * 16 + g + lh * 8;
            xw[(long long)row * N_ + ntile + lr] = o[g];
        }
    }
}

// ---------------------------------------------------------------------------
extern "C" void kernel_launch(void* const* d_in, const int* in_sizes, int n_in,
                              void* d_out, int out_size, void* d_ws, size_t ws_size,
                              hipStream_t stream) {
    const float* x      = (const float*)d_in[0];
    const float* lat    = (const float*)d_in[1];
    const float* w_q_l  = (const float*)d_in[2];
    const float* w_kv_x = (const float*)d_in[3];
    const float* w_q_x  = (const float*)d_in[4];
    const float* w_kv_l = (const float*)d_in[5];
    const float* w_proj = (const float*)d_in[6];
    const float* b_proj = (const float*)d_in[7];
    float* out = (float*)d_out;

    float* ws = (float*)d_ws;
    // workspace layout (floats); ~291 MB total
    float* W1   = ws;                       // 1536*512            = 786432
    float* qln  = W1   + 786432;            // 8*16*64             = 8192
    float* Y1   = qln  + 8192;              // 8*1536*4096         = 50331648
    float* rkx  = Y1   + 50331648;          // 64*4096             = 262144
    float* rqx  = rkx  + 262144;            // 262144
    float* Sbuf = rqx  + 262144;            // 64*16*4096          = 4194304
    float* mrow = Sbuf + 4194304;           // 1024
    float* rsro = mrow + 1024;              // 1024
    float* latr = rsro + 1024;              // 8*512*16            = 65536
    float* kvl  = latr + 65536;             // 8*1024*16           = 131072
    float* rkl  = kvl  + 131072;            // 1024
    float* xw   = rkl  + 1024;              // 8*512*4096          = 16777216

    // 0) weight concat + latent queries
    lm_concat_w<<<dim3((1536 * 512) / 256), 256, 0, stream>>>(w_kv_x, w_q_x, W1);
    lm_ql<<<dim3(128), 64, 0, stream>>>(w_q_l, lat, qln);

    // 1) Y1 = [w_kv_x; w_q_x] @ x[b]  (M=1536,N=4096,K=512, per batch)
    //    + fused k/q column-norm epilogue -> rkx, rqx
    lm_gemm_f32<<<dim3(64, 12, 8), 256, 0, stream>>>(
        W1, x, Y1, nullptr, rkx, rqx, 1536, N_, C_,
        (long long)C_ * N_, (long long)1536 * N_);

    // 2) attention 1 (latent read): scores -> stats -> output
    lm_attn1_scores<<<dim3(32, 64), 256, 0, stream>>>(Y1, qln, rkx, Sbuf);
    lm_attn1_stats<<<dim3(1024), 256, 0, stream>>>(Sbuf, mrow, rsro);
    lm_attn1_out<<<dim3(64), 128, 0, stream>>>(Y1, Sbuf, mrow, rsro, latr);

    // 3) kvl = w_kv_lat @ lat_read ; kl column norms
    lm_kvl<<<dim3(512), 256, 0, stream>>>(w_kv_l, latr, kvl);
    lm_rkl<<<dim3(1), 1024, 0, stream>>>(kvl, rkl);

    // 4) attention 2 (latent write) -> x_write
    lm_attn2<<<dim3(32, 64), 256, 0, stream>>>(Y1, kvl, rqx, rkl, xw);

    // 5) out = w_proj @ x_write + b_proj  (M=512,N=4096,K=512, per batch)
    lm_gemm_f32<<<dim3(64, 4, 8), 256, 0, stream>>>(
        w_proj, xw, out, b_proj, nullptr, nullptr, C_, N_, C_,
        (long long)C_ * N_, (long long)C_ * N_);
}